// SGCClassifier_30124900614171
// MI455X (gfx1250) — compile-verified
//
#include <hip/hip_runtime.h>
#include <hip/hip_bf16.h>
#include <math.h>

// ---- problem constants (match reference) ----
#define HID 1024
#define F1 512
#define F2 256
#define NCLS 40
#define NG 64
#define KPROP 5
#define BN_EPS 1e-5f

typedef float v2f __attribute__((ext_vector_type(2)));
typedef float v8f __attribute__((ext_vector_type(8)));

struct F3 { float x, y, z; };

static __device__ __forceinline__ unsigned fkey(float f) {
  unsigned u = __float_as_uint(f);
  return (u & 0x80000000u) ? ~u : (u | 0x80000000u);
}
static __device__ __forceinline__ float funkey(unsigned k) {
  unsigned u = (k & 0x80000000u) ? (k ^ 0x80000000u) : ~k;
  return __uint_as_float(u);
}

// ---------- Phase A: degree + rsqrt ----------
__global__ void k_degree(const int* __restrict__ col, float* __restrict__ deg, int E) {
  int e = blockIdx.x * blockDim.x + threadIdx.x;
  if (e >= E) return;
  // stream-prefetch the index array one chunk ahead (speculative, drop-safe)
  __builtin_prefetch(&col[min(e + 16384, E - 1)], 0, 0);
  atomicAdd(&deg[col[e]], 1.0f);
}

__global__ void k_rsqrt(float* __restrict__ dis, int N) {
  int i = blockIdx.x * blockDim.x + threadIdx.x;
  if (i >= N) return;
  dis[i] = rsqrtf(dis[i] + 1.0f);   // +1 self-loop; deg>=1 guaranteed
}

// ---------- Phase B: propagation x' = D^-1/2 (A+I) D^-1/2 x ----------
__global__ void k_selfloop(const float* __restrict__ dis, const float* __restrict__ xin,
                           float* __restrict__ xout, int N) {
  int i = blockIdx.x * blockDim.x + threadIdx.x;
  if (i >= N) return;
  float d = dis[i];
  float w = d * d;
  F3 v = ((const F3*)xin)[i];
  xout[3 * i + 0] = w * v.x;
  xout[3 * i + 1] = w * v.y;
  xout[3 * i + 2] = w * v.z;
}

__global__ void k_edge_prop(const int* __restrict__ row, const int* __restrict__ col,
                            const float* __restrict__ dis, const float* __restrict__ xin,
                            float* __restrict__ xout, int E) {
  int e = blockIdx.x * blockDim.x + threadIdx.x;
  if (e >= E) return;
  // keep the 2 index streams flowing from HBM ahead of use
  __builtin_prefetch(&row[min(e + 16384, E - 1)], 0, 0);
  __builtin_prefetch(&col[min(e + 16384, E - 1)], 0, 0);
  int r = row[e];
  int c = col[e];
  float w = dis[r] * dis[c];
  F3 v = ((const F3*)xin)[r];                 // single b96-class gather
  atomicAdd(&xout[3 * c + 0], w * v.x);
  atomicAdd(&xout[3 * c + 1], w * v.y);
  atomicAdd(&xout[3 * c + 2], w * v.z);
}

// ---------- Phase C1: sufficient statistics for BN0 (S1[3], S2[3x3]) ----------
__global__ void k_stats(const float* __restrict__ x, float* __restrict__ stats, int N) {
  __shared__ float sh[256 * 12];
  int t = threadIdx.x;
  float acc[12];
#pragma unroll
  for (int i = 0; i < 12; ++i) acc[i] = 0.f;
  for (int n = blockIdx.x * blockDim.x + t; n < N; n += gridDim.x * blockDim.x) {
    F3 v = ((const F3*)x)[n];
    acc[0] += v.x; acc[1] += v.y; acc[2] += v.z;
    acc[3] += v.x * v.x; acc[4] += v.x * v.y; acc[5] += v.x * v.z;
    acc[6] += v.y * v.x; acc[7] += v.y * v.y; acc[8] += v.y * v.z;
    acc[9] += v.z * v.x; acc[10] += v.z * v.y; acc[11] += v.z * v.z;
  }
#pragma unroll
  for (int i = 0; i < 12; ++i) sh[t * 12 + i] = acc[i];
  __syncthreads();
  for (int s = 128; s > 0; s >>= 1) {
    if (t < s) {
#pragma unroll
      for (int i = 0; i < 12; ++i) sh[t * 12 + i] += sh[(t + s) * 12 + i];
    }
    __syncthreads();
  }
  if (t < 12) atomicAdd(&stats[t], sh[t]);
}

// ---------- Phase C2: fused Linear(3->1024) + per-graph running max/min ----------
// grid.x = node chunks of 1024, grid.y = 8 channel tiles of 128; block = 128
__global__ void k_linmax(const float* __restrict__ x, const int* __restrict__ batch,
                         const float* __restrict__ W, const float* __restrict__ bias,
                         unsigned* __restrict__ gmaxk, unsigned* __restrict__ gmink, int N) {
  int j = blockIdx.y * 128 + threadIdx.x;      // channel 0..1023
  int n0 = blockIdx.x * 1024;
  int n1 = min(N, n0 + 1024);
  if (n0 >= n1) return;
  float w0 = W[0 * HID + j], w1 = W[1 * HID + j], w2 = W[2 * HID + j], bj = bias[j];
  float cmax = -3.402823466e38f, cmin = 3.402823466e38f;
  int curg = batch[n0];
  for (int n = n0; n < n1; ++n) {
    int g = batch[n];
    if (g != curg) {
      atomicMax(&gmaxk[curg * HID + j], fkey(cmax));
      atomicMin(&gmink[curg * HID + j], fkey(cmin));
      curg = g;
      cmax = -3.402823466e38f;
      cmin = 3.402823466e38f;
    }
    F3 v = ((const F3*)x)[n];                  // broadcast hit in WGP$/L2
    float y = fmaf(v.z, w2, fmaf(v.y, w1, fmaf(v.x, w0, bj)));
    cmax = fmaxf(cmax, y);
    cmin = fminf(cmin, y);
  }
  atomicMax(&gmaxk[curg * HID + j], fkey(cmax));
  atomicMin(&gmink[curg * HID + j], fkey(cmin));
}

// ---------- Phase C3: BN0 coefficients from stats + finalize pooled ----------
__global__ void k_bn0_pool(const float* __restrict__ stats, const unsigned* __restrict__ gmaxk,
                           const unsigned* __restrict__ gmink, const float* __restrict__ W,
                           const float* __restrict__ bias, const float* __restrict__ g0,
                           const float* __restrict__ b0, float* __restrict__ pooled, int N) {
  int j = blockIdx.x * blockDim.x + threadIdx.x;
  if (j >= HID) return;
  float w0 = W[0 * HID + j], w1 = W[1 * HID + j], w2 = W[2 * HID + j], bj = bias[j];
  float invN = 1.0f / (float)N;
  float s1w = stats[0] * w0 + stats[1] * w1 + stats[2] * w2;
  float mean = s1w * invN + bj;
  const float* S2 = stats + 3;
  float q = w0 * (S2[0] * w0 + S2[1] * w1 + S2[2] * w2) +
            w1 * (S2[3] * w0 + S2[4] * w1 + S2[5] * w2) +
            w2 * (S2[6] * w0 + S2[7] * w1 + S2[8] * w2);
  float ex2 = q * invN + 2.f * bj * s1w * invN + bj * bj;
  float var = ex2 - mean * mean;
  float a = g0[j] * rsqrtf(var + BN_EPS);
  float c = b0[j] - a * mean;
  for (int g = 0; g < NG; ++g) {
    float vmax = funkey(gmaxk[g * HID + j]);
    float vmin = funkey(gmink[g * HID + j]);
    float v = (a >= 0.f) ? vmax : vmin;   // monotone affine: max commutes through BN
    pooled[g * HID + j] = fmaxf(fmaf(a, v, c), 0.f);
  }
}

// ---------- Phase D: WMMA f32 GEMM (C = A[MxK] * B[KxN] + bias) ----------
// One wave computes one 16x16 tile via V_WMMA_F32_16X16X4_F32 over a K/4 loop.
// N, K are compile-time so the backend folds B's row stride into immediates.
template <int N, int K>
__global__ void k_gemm_wmma(const float* __restrict__ A, const float* __restrict__ B,
                            const float* __restrict__ bias, float* __restrict__ C,
                            int M) {
  int wid = (blockIdx.x * blockDim.x + threadIdx.x) >> 5;
  int lane = threadIdx.x & 31;
  constexpr int tilesN = N >> 4;
  int tm = wid / tilesN;
  int tn = wid % tilesN;
  if (tm * 16 >= M) return;                    // wave-uniform guard (EXEC stays all-ones)
  int mrow = tm * 16 + (lane & 15);
  int ncol = tn * 16 + (lane & 15);
  int ko = (lane & 16) ? 2 : 0;                // lanes 16..31 carry K=2,3
  const float* Ap = A + (size_t)mrow * K + ko;
  const float* Bp = B + (size_t)ko * N + ncol;
  v8f acc = {};
#pragma unroll 8
  for (int k = 0; k < K; k += 4) {
    v2f a, b;
    a.x = Ap[0];
    a.y = Ap[1];
    b.x = Bp[0];
    b.y = Bp[N];
    Ap += 4;
    Bp += 4 * N;
    acc = __builtin_amdgcn_wmma_f32_16x16x4_f32(false, a, false, b, (short)0, acc,
                                                false, false);
  }
  int mbase = tm * 16 + ((lane & 16) ? 8 : 0); // C/D layout: lanes 16..31 hold M=8..15
  int n = tn * 16 + (lane & 15);
  float bv = bias[n];
#pragma unroll
  for (int r = 0; r < 8; ++r) C[(size_t)(mbase + r) * N + n] = acc[r] + bv;
}

// ---------- BN over rows (64) + ReLU, in place ----------
__global__ void k_bn_relu_rows(float* __restrict__ x, const float* __restrict__ g,
                               const float* __restrict__ b, int rows, int F) {
  int j = blockIdx.x * blockDim.x + threadIdx.x;
  if (j >= F) return;
  float s = 0.f, s2 = 0.f;
  for (int r = 0; r < rows; ++r) {
    float v = x[(size_t)r * F + j];
    s += v;
    s2 += v * v;
  }
  float inv = 1.0f / (float)rows;
  float mean = s * inv;
  float var = s2 * inv - mean * mean;
  float a = g[j] * rsqrtf(var + BN_EPS);
  float c = b[j] - a * mean;
  for (int r = 0; r < rows; ++r)
    x[(size_t)r * F + j] = fmaxf(fmaf(a, x[(size_t)r * F + j], c), 0.f);
}

// ---------- Head: 64x256 @ 256x40 + log_softmax ----------
__global__ void k_head(const float* __restrict__ h, const float* __restrict__ W,
                       const float* __restrict__ b, float* __restrict__ out) {
  __shared__ float sl[64];
  __shared__ float red[2];
  int g = blockIdx.x, t = threadIdx.x;
  float z = -3.402823466e38f;
  if (t < NCLS) {
    z = b[t];
    for (int k = 0; k < F2; ++k) z = fmaf(h[g * F2 + k], W[k * NCLS + t], z);
  }
  sl[t] = z;
  __syncthreads();
  if (t == 0) {
    float m = sl[0];
    for (int i = 1; i < NCLS; ++i) m = fmaxf(m, sl[i]);
    float s = 0.f;
    for (int i = 0; i < NCLS; ++i) s += expf(sl[i] - m);
    red[0] = m;
    red[1] = logf(s);
  }
  __syncthreads();
  if (t < NCLS) out[g * NCLS + t] = z - red[0] - red[1];
}

static inline int cdiv(int a, int b) { return (a + b - 1) / b; }

extern "C" void kernel_launch(void* const* d_in, const int* in_sizes, int n_in,
                              void* d_out, int out_size, void* d_ws, size_t ws_size,
                              hipStream_t stream) {
  const float* pos   = (const float*)d_in[0];
  const int*   ei    = (const int*)d_in[1];
  const int*   batch = (const int*)d_in[2];
  const float* lin_w = (const float*)d_in[3];
  const float* lin_b = (const float*)d_in[4];
  const float* bn0_g = (const float*)d_in[5];
  const float* bn0_b = (const float*)d_in[6];
  const float* fc1_w = (const float*)d_in[7];
  const float* fc1_b = (const float*)d_in[8];
  const float* bn1_g = (const float*)d_in[9];
  const float* bn1_b = (const float*)d_in[10];
  const float* fc2_w = (const float*)d_in[11];
  const float* fc2_b = (const float*)d_in[12];
  const float* bn2_g = (const float*)d_in[13];
  const float* bn2_b = (const float*)d_in[14];
  const float* fc3_w = (const float*)d_in[15];
  const float* fc3_b = (const float*)d_in[16];
  float* out = (float*)d_out;

  const int N = in_sizes[0] / 3;
  const int E = in_sizes[1] / 2;
  const int* row = ei;
  const int* col = ei + E;

  // ---- workspace carve-up ----
  char* base = (char*)d_ws;
  size_t off = 0;
  auto carve = [&](size_t bytes) -> char* {
    char* p = base + off;
    off = (off + bytes + 255) & ~(size_t)255;
    return p;
  };
  float*    dis    = (float*)carve((size_t)N * 4);
  float*    x0     = (float*)carve((size_t)N * 3 * 4);
  float*    x1     = (float*)carve((size_t)N * 3 * 4);
  float*    stats  = (float*)carve(12 * 4);
  unsigned* gmaxk  = (unsigned*)carve((size_t)NG * HID * 4);
  unsigned* gmink  = (unsigned*)carve((size_t)NG * HID * 4);
  float*    pooled = (float*)carve((size_t)NG * HID * 4);
  float*    h1     = (float*)carve((size_t)NG * F1 * 4);
  float*    h2     = (float*)carve((size_t)NG * F2 * 4);
  (void)ws_size;

  // ---- Phase A: degree / normalization ----
  hipMemsetAsync(dis, 0, (size_t)N * 4, stream);
  k_degree<<<cdiv(E, 256), 256, 0, stream>>>(col, dis, E);
  k_rsqrt<<<cdiv(N, 256), 256, 0, stream>>>(dis, N);

  // ---- Phase B: K=5 propagation sweeps (ping-pong) ----
  const float* src = pos;
  float* dst = x0;
  for (int it = 0; it < KPROP; ++it) {
    k_selfloop<<<cdiv(N, 256), 256, 0, stream>>>(dis, src, dst, N);
    k_edge_prop<<<cdiv(E, 256), 256, 0, stream>>>(row, col, dis, src, dst, E);
    src = dst;
    dst = (dst == x0) ? x1 : x0;
  }

  // ---- Phase C: stats + fused linear/segment-max + BN0 pooled ----
  hipMemsetAsync(stats, 0, 12 * 4, stream);
  k_stats<<<64, 256, 0, stream>>>(src, stats, N);
  hipMemsetAsync(gmaxk, 0x00, (size_t)NG * HID * 4, stream);
  hipMemsetAsync(gmink, 0xFF, (size_t)NG * HID * 4, stream);
  {
    dim3 grid(cdiv(N, 1024), HID / 128);
    k_linmax<<<grid, 128, 0, stream>>>(src, batch, lin_w, lin_b, gmaxk, gmink, N);
  }
  k_bn0_pool<<<HID / 128, 128, 0, stream>>>(stats, gmaxk, gmink, lin_w, lin_b,
                                            bn0_g, bn0_b, pooled, N);

  // ---- Phase D: MLP head with f32 WMMA ----
  // GEMM1: 64x1024 @ 1024x512 -> 4*32 = 128 tiles (waves), 4 waves/block
  k_gemm_wmma<F1, HID><<<(NG / 16) * (F1 / 16) / 4, 128, 0, stream>>>(pooled, fc1_w,
                                                                      fc1_b, h1, NG);
  k_bn_relu_rows<<<cdiv(F1, 128), 128, 0, stream>>>(h1, bn1_g, bn1_b, NG, F1);
  // GEMM2: 64x512 @ 512x256 -> 4*16 = 64 tiles
  k_gemm_wmma<F2, F1><<<(NG / 16) * (F2 / 16) / 4, 128, 0, stream>>>(h1, fc2_w,
                                                                     fc2_b, h2, NG);
  k_bn_relu_rows<<<cdiv(F2, 128), 128, 0, stream>>>(h2, bn2_g, bn2_b, NG, F2);
  k_head<<<NG, 64, 0, stream>>>(h2, fc3_w, fc3_b, out);
}